// NonSpikingInterneuron_69509750719021
// MI455X (gfx1250) — compile-verified
//
#include <hip/hip_runtime.h>

typedef __attribute__((ext_vector_type(2))) float v2f;
typedef __attribute__((ext_vector_type(4))) float v4f;
typedef __attribute__((ext_vector_type(8))) float v8f;

#ifndef TB
#define TB 64
#endif

// Reference RK2 step collapses to: v' = a*v + b, a = aC - s*g, b = s*E*g,
// g = g0[r,t] + g1[c,t].  With w = v - E:  w' = a*w + wK  (one FMA/elem/step).
// WMMA (16x16x4, f32) builds the full 'a' tile per step as THREE rank-1 terms
// using K=0..2, so SRC2 lowers to inline 0 (no C register copies):
//   K=0: (-s*g0) (x) 1     K=1: 1 (x) (-s*g1)     K=2: 1 (x) aC      K=3: 0
__global__ __launch_bounds__(32)
void nsi_rk2_wmma(const float* __restrict__ u0,
                  const float* __restrict__ u1,
                  const float* __restrict__ ksyn,
                  float* __restrict__ out,
                  int N, int T)
{
    constexpr float DT = 0.1f, C_M = 1.0f, E_REST = 0.0f, G_M_LEAK = 5.0f,
                    E_SYN = 20.0f, R_OP = 1.0f;
    const float E  = E_REST + E_SYN;                            // 20
    const float s  = DT * (1.0f - 0.5f * DT * G_M_LEAK / C_M);  // 0.075
    const float aC = 1.0f - s * G_M_LEAK / C_M;                 // 0.625
    const float wK = -s * G_M_LEAK * E / C_M;                   // -7.5

    const int lane   = threadIdx.x;          // one wave32 per block
    const int lane16 = lane & 15;
    const bool lo    = lane < 16;
    const int nT16   = N >> 4;
    const int r0     = (blockIdx.x / nT16) << 4;
    const int c0     = (blockIdx.x % nT16) << 4;

    __shared__ float gm[32];                         // -s*gmax for rows / cols
    __shared__ __align__(16) float g0s[16][TB + 4];  // +4 pad: conflict-free b128 reads AND
    __shared__ __align__(16) float g1s[16][TB + 4];  //         legal pipelined overread at j4+4

    {   // -s * k/(E - 2k)
        int idx   = lo ? (r0 + lane) : (N + c0 + lane16);
        float k   = ksyn[idx];
        float num = k * R_OP;
        gm[lane]  = -s * (num / (E - 2.0f * num));
    }
    __syncthreads();

    float  w[8];
    float* base[8];
#pragma unroll
    for (int k = 0; k < 8; ++k) {
        w[k]  = E_REST - E;                          // v0 = E_REST -> w0 = -20
        int M = k + ((lane >> 4) << 3);              // C/D layout: VGPR k -> M=k / M=k+8
        base[k] = out + ((size_t)(r0 + M) * N + (c0 + lane16)) * T;
    }

    const float aOne = lo ? 1.0f : 0.0f;   // A vgpr1: K=1 col ones, K=3 col zero
    const float bOne = lo ? 1.0f : aC;     // B vgpr0: K=0 row ones, K=2 row aC
    const bool vecOK = (T & 3) == 0;
    const float invR = 1.0f / R_OP;

    for (int t0 = 0; t0 < T; t0 += TB) {
        // ---- stage -s*g chunk into LDS ----
        if (vecOK && t0 + TB <= T) {                 // fast path: b128 everywhere
            const int row = lane >> 4;               // 0..1
            const int col = lane16 << 2;             // 0,4,...,60
#pragma unroll
            for (int i = 0; i < 16; i += 2) {
                int r = i + row;
                float sA = gm[r], sB = gm[16 + r];
                v4f x0 = *(const v4f*)(u0 + (size_t)(r0 + r) * T + t0 + col);
                v4f x1 = *(const v4f*)(u1 + (size_t)(c0 + r) * T + t0 + col);
#pragma unroll
                for (int q = 0; q < 4; ++q) {
                    x0[q] = sA * fminf(fmaxf(x0[q] * invR, 0.0f), 1.0f);
                    x1[q] = sB * fminf(fmaxf(x1[q] * invR, 0.0f), 1.0f);
                }
                *(v4f*)&g0s[r][col] = x0;
                *(v4f*)&g1s[r][col] = x1;
            }
        } else {                                     // tail/unaligned fallback
            for (int i = 0; i < 16; ++i)
                for (int j = lane; j < TB; j += 32) {
                    int t = t0 + j;
                    float a0 = 0.0f, a1 = 0.0f;
                    if (t < T) {
                        float x0 = u0[(size_t)(r0 + i) * T + t] * invR;
                        a0 = gm[i] * fminf(fmaxf(x0, 0.0f), 1.0f);
                        float x1 = u1[(size_t)(c0 + i) * T + t] * invR;
                        a1 = gm[16 + i] * fminf(fmaxf(x1, 0.0f), 1.0f);
                    }
                    g0s[i][j] = a0;
                    g1s[i][j] = a1;
                }
        }
        __syncthreads();

        const int tEnd  = (T - t0 < TB) ? (T - t0) : TB;
        const int tFull = vecOK ? (tEnd & ~3) : 0;   // !vecOK: all steps via scalar tail

        // ---- steady state: branch-free, software-pipelined 4-step groups ----
        if (tFull > 0) {
            v4f c0v = *(const v4f*)&g0s[lane16][0];
            v4f c1v = *(const v4f*)&g1s[lane16][0];
#pragma unroll 2
            for (int j4 = 0; j4 < tFull; j4 += 4) {
                // prefetch next group's g values (pad makes j4+4==TB overread legal)
                v4f n0 = *(const v4f*)&g0s[lane16][j4 + 4];
                v4f n1 = *(const v4f*)&g1s[lane16][j4 + 4];
                v4f buf[8];
#pragma unroll
                for (int q = 0; q < 4; ++q) {
                    v2f A; A[0] = lo ? c0v[q] : 1.0f; A[1] = aOne;
                    v2f B; B[0] = bOne;               B[1] = c1v[q]; // hi-lane garbage killed by A K=3 = 0
                    v8f Cz = {};                                     // -> inline SRC2 = 0
                    v8f at = __builtin_amdgcn_wmma_f32_16x16x4_f32(
                        false, A, false, B, (short)0, Cz, false, false);
#pragma unroll
                    for (int k = 0; k < 8; ++k) {
                        w[k]      = __builtin_fmaf(at[k], w[k], wK);
                        buf[k][q] = w[k] + E;                        // v = w + 20
                    }
                }
#pragma unroll
                for (int k = 0; k < 8; ++k)    // b128 streaming stores; 192MB L2 merges lines
                    *(v4f*)(base[k] + t0 + j4) = buf[k];
                c0v = n0; c1v = n1;
            }
        }

        // ---- scalar tail (remainder steps, or whole chunk when !vecOK) ----
        for (int j = tFull; j < tEnd; ++j) {
            float g0v = g0s[lane16][j];
            float g1v = g1s[lane16][j];
            v2f A; A[0] = lo ? g0v : 1.0f; A[1] = aOne;
            v2f B; B[0] = bOne;            B[1] = g1v;
            v8f Cz = {};
            v8f at = __builtin_amdgcn_wmma_f32_16x16x4_f32(
                false, A, false, B, (short)0, Cz, false, false);
#pragma unroll
            for (int k = 0; k < 8; ++k) {
                w[k] = __builtin_fmaf(at[k], w[k], wK);
                base[k][t0 + j] = w[k] + E;
            }
        }
        __syncthreads();
    }
}

extern "C" void kernel_launch(void* const* d_in, const int* in_sizes, int n_in,
                              void* d_out, int out_size, void* d_ws, size_t ws_size,
                              hipStream_t stream)
{
    (void)n_in; (void)out_size; (void)d_ws; (void)ws_size;
    const float* u0 = (const float*)d_in[0];   // u_pre_0 (N,T) f32
    const float* u1 = (const float*)d_in[1];   // u_pre_1 (N,T) f32
    const float* ks = (const float*)d_in[2];   // k_syn   (2,N) f32
    float* out = (float*)d_out;                // (N,N,T) f32

    const int N = in_sizes[2] / 2;             // 256
    const int T = in_sizes[0] / N;             // 1500
    const int nTiles = (N / 16) * (N / 16);    // one wave per 16x16 output tile

    nsi_rk2_wmma<<<dim3(nTiles), dim3(32), 0, stream>>>(u0, u1, ks, out, N, T);
}